// MulticlassDecorrelatedLoss_36206574305497
// MI455X (gfx1250) — compile-verified
//
#include <hip/hip_runtime.h>
#include <math.h>

// ---------------------------------------------------------------------------
// MulticlassDecorrelatedLoss for MI455X (gfx1250, wave32)
//
// Math: weighted distance correlation with doubly-centered matrices reduces to
//   S_AB = s^2*Σω R  - (2 s^3/N)*Σω A B + (s^4/N^2)*(Σω A)(Σω B)
//   S_AA = 2Ns Σω p^2 - 2 s^2 (Σω p)^2 - (2 s^3/N) Σω A^2 + (s^4/N^2)(Σω A)^2
// where A[i]=Σ_j ω_j|p_i-p_j|, B[i]=Σ_j ω_j|e_i-e_j|, R[i]=Σ_j ω_j|Δp||Δe|,
// s = N/Σω.  Only A,B,R are O(N^2); A,B row sums are exact f32 mat×vec
// contractions mapped onto V_WMMA_F32_16X16X4_F32 (matrix pipe co-executes
// with the VALU |Δ| work).  R stays VALU fma.  All final reductions are
// fixed-order (deterministic), done in double in a single workgroup.
// Workspace: 41*N floats (~1.35 MB).
// ---------------------------------------------------------------------------

typedef __attribute__((ext_vector_type(2))) float v2f;
typedef __attribute__((ext_vector_type(8))) float v8f;

static constexpr int kN      = 8192;
static constexpr int kC      = 4;
static constexpr int kSplit  = 4;              // j-range splits
static constexpr int kJChunk = kN / kSplit;    // 2048 j per workgroup
static constexpr int kIBlock = 128;            // 8 waves * 16 rows
static constexpr int kNArr   = 2 * kC + 1;     // A0..A3, B, R0..R3

#if __has_builtin(__builtin_amdgcn_wmma_f32_16x16x4_f32)
#define USE_WMMA 1
#else
#define USE_WMMA 0
#endif

// ------------------------- Kernel 0: softmax + CE --------------------------
__global__ __launch_bounds__(256) void k_softmax_ce(
    const float* __restrict__ outputs, const float* __restrict__ w,
    const int* __restrict__ labels, float* __restrict__ probs,
    float* __restrict__ ce)
{
  const int i = blockIdx.x * 256 + threadIdx.x;
  if (i >= kN) return;
  const float x0 = outputs[4 * i + 0], x1 = outputs[4 * i + 1];
  const float x2 = outputs[4 * i + 2], x3 = outputs[4 * i + 3];
  const float m  = fmaxf(fmaxf(x0, x1), fmaxf(x2, x3));
  const float e0 = expf(x0 - m), e1 = expf(x1 - m);
  const float e2 = expf(x2 - m), e3 = expf(x3 - m);
  const float z   = e0 + e1 + e2 + e3;
  const float inv = 1.0f / z;
  const float lz  = logf(z);
  probs[0 * kN + i] = e0 * inv;
  probs[1 * kN + i] = e1 * inv;
  probs[2 * kN + i] = e2 * inv;
  probs[3 * kN + i] = e3 * inv;
  const int lab = labels[i];
  const float xl = (lab == 0) ? x0 : (lab == 1) ? x1 : (lab == 2) ? x2 : x3;
  ce[i] = -(xl - m - lz) * w[i];
}

// --------------------- Kernel 1: O(N^2) pairwise row sums ------------------
// grid = (kN/kIBlock, kSplit), block = 256 (8 waves, 16 rows each).
__global__ __launch_bounds__(256) void k_pairwise(
    const float* __restrict__ probs, const float* __restrict__ event,
    const float* __restrict__ w, float* __restrict__ partials)
{
  __shared__ float lds[6 * kJChunk];  // e | w | p0 | p1 | p2 | p3  (48 KB)
  const int tid   = threadIdx.x;
  const int split = blockIdx.y;
  const int j0    = split * kJChunk;

  for (int k = tid; k < kJChunk; k += 256) {
    lds[0 * kJChunk + k] = event[j0 + k];
    lds[1 * kJChunk + k] = w[j0 + k];
    lds[2 * kJChunk + k] = probs[0 * kN + j0 + k];
    lds[3 * kJChunk + k] = probs[1 * kN + j0 + k];
    lds[4 * kJChunk + k] = probs[2 * kN + j0 + k];
    lds[5 * kJChunk + k] = probs[3 * kN + j0 + k];
  }
  __syncthreads();

  const int lane = tid & 31;
  const int wave = tid >> 5;
  const int half = lane >> 4;   // 0 -> K{0,1}, 1 -> K{2,3}  (WMMA A layout)
  const int row  = lane & 15;   // matrix row M
  const int i    = blockIdx.x * kIBlock + wave * 16 + row;

  const float ei  = event[i];
  const float pi0 = probs[0 * kN + i];
  const float pi1 = probs[1 * kN + i];
  const float pi2 = probs[2 * kN + i];
  const float pi3 = probs[3 * kN + i];

#if USE_WMMA
  v8f accB = 0.0f, acc0 = 0.0f, acc1 = 0.0f, acc2 = 0.0f, acc3 = 0.0f;
#else
  float accB = 0.f, acc0 = 0.f, acc1 = 0.f, acc2 = 0.f, acc3 = 0.f;
#endif
  float R0 = 0.f, R1 = 0.f, R2 = 0.f, R3 = 0.f;
  const int joff = half * 2;

  for (int t = 0; t < kJChunk; t += 4) {
    const int jA = t + joff;  // even -> 8B-aligned float2 LDS loads
    const float2 ee = *reinterpret_cast<const float2*>(&lds[0 * kJChunk + jA]);
    const float2 uu = *reinterpret_cast<const float2*>(&lds[1 * kJChunk + jA]);
    const float2 q0 = *reinterpret_cast<const float2*>(&lds[2 * kJChunk + jA]);
    const float2 q1 = *reinterpret_cast<const float2*>(&lds[3 * kJChunk + jA]);
    const float2 q2 = *reinterpret_cast<const float2*>(&lds[4 * kJChunk + jA]);
    const float2 q3 = *reinterpret_cast<const float2*>(&lds[5 * kJChunk + jA]);

    const float beA = fabsf(ei - ee.x), beB = fabsf(ei - ee.y);
    const float tA  = beA * uu.x,       tB  = beB * uu.y;

    const float a0A = fabsf(pi0 - q0.x), a0B = fabsf(pi0 - q0.y);
    const float a1A = fabsf(pi1 - q1.x), a1B = fabsf(pi1 - q1.y);
    const float a2A = fabsf(pi2 - q2.x), a2B = fabsf(pi2 - q2.y);
    const float a3A = fabsf(pi3 - q3.x), a3B = fabsf(pi3 - q3.y);

#if USE_WMMA
    const v2f bu = {uu.x, uu.y};
    {
      v2f fa = {beA, beB};
      accB = __builtin_amdgcn_wmma_f32_16x16x4_f32(false, fa, false, bu,
                                                   (short)0, accB, false, false);
    }
    {
      v2f fa = {a0A, a0B};
      acc0 = __builtin_amdgcn_wmma_f32_16x16x4_f32(false, fa, false, bu,
                                                   (short)0, acc0, false, false);
    }
    {
      v2f fa = {a1A, a1B};
      acc1 = __builtin_amdgcn_wmma_f32_16x16x4_f32(false, fa, false, bu,
                                                   (short)0, acc1, false, false);
    }
    {
      v2f fa = {a2A, a2B};
      acc2 = __builtin_amdgcn_wmma_f32_16x16x4_f32(false, fa, false, bu,
                                                   (short)0, acc2, false, false);
    }
    {
      v2f fa = {a3A, a3B};
      acc3 = __builtin_amdgcn_wmma_f32_16x16x4_f32(false, fa, false, bu,
                                                   (short)0, acc3, false, false);
    }
#else
    accB = fmaf(beA, uu.x, fmaf(beB, uu.y, accB));
    acc0 = fmaf(a0A, uu.x, fmaf(a0B, uu.y, acc0));
    acc1 = fmaf(a1A, uu.x, fmaf(a1B, uu.y, acc1));
    acc2 = fmaf(a2A, uu.x, fmaf(a2B, uu.y, acc2));
    acc3 = fmaf(a3A, uu.x, fmaf(a3B, uu.y, acc3));
#endif
    R0 = fmaf(a0A, tA, fmaf(a0B, tB, R0));
    R1 = fmaf(a1A, tA, fmaf(a1B, tB, R1));
    R2 = fmaf(a2A, tA, fmaf(a2B, tB, R2));
    R3 = fmaf(a3A, tA, fmaf(a3B, tB, R3));
  }

  // lane L and L^16 covered complementary j columns of the same row i.
  R0 += __shfl_xor(R0, 16, 32);
  R1 += __shfl_xor(R1, 16, 32);
  R2 += __shfl_xor(R2, 16, 32);
  R3 += __shfl_xor(R3, 16, 32);

  float* PB = partials + (size_t)split * kNArr * kN;
  if (half == 0) {
    const int ii = blockIdx.x * kIBlock + wave * 16 + row;
    PB[(5 + 0) * kN + ii] = R0;
    PB[(5 + 1) * kN + ii] = R1;
    PB[(5 + 2) * kN + ii] = R2;
    PB[(5 + 3) * kN + ii] = R3;
  }

#if USE_WMMA
  // D layout: lanes 0-15 vgpr r = row r; lanes 16-31 vgpr r = row 8+r.
  // All 16 D columns are identical (B columns identical), so lanes 0 and 16
  // of each wave hold the 16 row sums.
  if (row == 0) {
    const int base = blockIdx.x * kIBlock + wave * 16 + half * 8;
#pragma unroll
    for (int r = 0; r < 8; r++) {
      PB[0 * kN + base + r] = acc0[r];
      PB[1 * kN + base + r] = acc1[r];
      PB[2 * kN + base + r] = acc2[r];
      PB[3 * kN + base + r] = acc3[r];
      PB[4 * kN + base + r] = accB[r];
    }
  }
#else
  acc0 += __shfl_xor(acc0, 16, 32);
  acc1 += __shfl_xor(acc1, 16, 32);
  acc2 += __shfl_xor(acc2, 16, 32);
  acc3 += __shfl_xor(acc3, 16, 32);
  accB += __shfl_xor(accB, 16, 32);
  if (half == 0) {
    const int ii = blockIdx.x * kIBlock + wave * 16 + row;
    PB[0 * kN + ii] = acc0;
    PB[1 * kN + ii] = acc1;
    PB[2 * kN + ii] = acc2;
    PB[3 * kN + ii] = acc3;
    PB[4 * kN + ii] = accB;
  }
#endif
}

// ------------------- Kernel 2: deterministic finalization ------------------
__global__ __launch_bounds__(256) void k_finalize(
    const float* __restrict__ probs, const float* __restrict__ event,
    const float* __restrict__ w, const float* __restrict__ ce,
    const float* __restrict__ partials, float* __restrict__ out3)
{
  // sums: 0 Σω | 1 Σce | 2 Σωe | 3 Σωe² | 4 ΣωB | 5 ΣωB²
  // per class c (base 6+6c): Σωp, Σωp², ΣωA, ΣωA², ΣωAB, ΣωR
  double loc[30];
#pragma unroll
  for (int q = 0; q < 30; q++) loc[q] = 0.0;

  for (int i = threadIdx.x; i < kN; i += 256) {
    const float wi = w[i], ev = event[i], cei = ce[i];
    float Bv = 0.f, Av[4] = {0.f, 0.f, 0.f, 0.f}, Rv[4] = {0.f, 0.f, 0.f, 0.f};
    for (int s = 0; s < kSplit; s++) {
      const float* PB = partials + (size_t)s * kNArr * kN;
      Av[0] += PB[0 * kN + i];
      Av[1] += PB[1 * kN + i];
      Av[2] += PB[2 * kN + i];
      Av[3] += PB[3 * kN + i];
      Bv    += PB[4 * kN + i];
      Rv[0] += PB[5 * kN + i];
      Rv[1] += PB[6 * kN + i];
      Rv[2] += PB[7 * kN + i];
      Rv[3] += PB[8 * kN + i];
    }
    loc[0] += wi;
    loc[1] += cei;
    loc[2] += (double)wi * ev;
    loc[3] += (double)wi * ev * ev;
    loc[4] += (double)wi * Bv;
    loc[5] += (double)wi * Bv * Bv;
    for (int c = 0; c < kC; c++) {
      const float p = probs[c * kN + i];
      const int b = 6 + 6 * c;
      loc[b + 0] += (double)wi * p;
      loc[b + 1] += (double)wi * p * p;
      loc[b + 2] += (double)wi * Av[c];
      loc[b + 3] += (double)wi * Av[c] * Av[c];
      loc[b + 4] += (double)wi * Av[c] * Bv;
      loc[b + 5] += (double)wi * Rv[c];
    }
  }

  __shared__ double red[256];
  __shared__ double sums[30];
  for (int q = 0; q < 30; q++) {
    red[threadIdx.x] = loc[q];
    __syncthreads();
    for (int off = 128; off > 0; off >>= 1) {
      if (threadIdx.x < off) red[threadIdx.x] += red[threadIdx.x + off];
      __syncthreads();
    }
    if (threadIdx.x == 0) sums[q] = red[0];
    __syncthreads();
  }

  if (threadIdx.x == 0) {
    const double Nn = (double)kN;
    const double W  = sums[0];
    const double s  = Nn / W;
    const double s2 = s * s, s3 = s2 * s, s4 = s2 * s2;

    const double SBe = sums[2], SBe2 = sums[3], SB = sums[4], SB2 = sums[5];
    const double T1bb = 2.0 * Nn * s * SBe2 - 2.0 * s2 * SBe * SBe;
    const double S_BB =
        T1bb - (2.0 * s3 / Nn) * SB2 + (s4 / (Nn * Nn)) * SB * SB;

    double disco = 0.0;
    for (int c = 0; c < kC; c++) {
      const int b = 6 + 6 * c;
      const double Sp = sums[b], Sp2 = sums[b + 1], SA = sums[b + 2];
      const double SA2 = sums[b + 3], SAB = sums[b + 4], SR = sums[b + 5];
      const double T1aa = 2.0 * Nn * s * Sp2 - 2.0 * s2 * Sp * Sp;
      const double S_AA =
          T1aa - (2.0 * s3 / Nn) * SA2 + (s4 / (Nn * Nn)) * SA * SA;
      const double S_AB =
          s2 * SR - (2.0 * s3 / Nn) * SAB + (s4 / (Nn * Nn)) * SA * SB;
      disco += S_AB / sqrt(S_AA * S_BB);
    }
    disco *= (1.0 / (double)kC);
    const double cem = sums[1] / Nn;
    out3[0] = (float)cem;
    out3[1] = (float)disco;
    out3[2] = (float)(cem + 0.1 * disco);
  }
}

// ---------------------------------------------------------------------------
extern "C" void kernel_launch(void* const* d_in, const int* in_sizes, int n_in,
                              void* d_out, int out_size, void* d_ws,
                              size_t ws_size, hipStream_t stream) {
  (void)in_sizes; (void)n_in; (void)out_size; (void)ws_size;
  const float* outputs = (const float*)d_in[0];
  const float* event   = (const float*)d_in[1];
  const float* weights = (const float*)d_in[2];
  const int*   labels  = (const int*)d_in[3];

  float* ws       = (float*)d_ws;
  float* probs    = ws;                 // 4N floats (SoA by class)
  float* ce       = ws + 4 * kN;        // N floats
  float* partials = ws + 5 * kN;        // kSplit * 9 * N floats
  float* out3     = (float*)d_out;

  k_softmax_ce<<<kN / 256, 256, 0, stream>>>(outputs, weights, labels, probs,
                                             ce);
  dim3 g1(kN / kIBlock, kSplit);
  k_pairwise<<<g1, 256, 0, stream>>>(probs, event, weights, partials);
  k_finalize<<<1, 256, 0, stream>>>(probs, event, weights, ce, partials, out3);
}